// PesqLoss_18056042512851
// MI455X (gfx1250) — compile-verified
//
#include <hip/hip_runtime.h>
#include <math.h>

#define T_SAMP 160000
#define NB 64
#define NFRAMES 624
#define NBARK 49
#define NFFT 512
#define LCH 500
#define NCH 320            // 160000 / 500
#define TOTF (NB*NFRAMES)  // 39936

#define C_B0 2.740826f
#define C_B1 (-5.4816519f)
#define C_B2 2.740826f
#define C_A1 (-1.9444777f)
#define C_A2 0.94597794f
#define PI_D 3.14159265358979323846

typedef float v2f __attribute__((ext_vector_type(2)));
typedef float v8f __attribute__((ext_vector_type(8)));

// ---------------------------------------------------------------- tables ----
__global__ void k_init_tables(float* __restrict__ win, int* __restrict__ band,
                              float* __restrict__ pc, float* __restrict__ th,
                              float* __restrict__ gS) {
  int t = threadIdx.x;
  if (t < NFFT) win[t] = (float)(0.5 - 0.5 * cos(2.0 * PI_D * (double)t / 512.0));
  if (t < 256) {
    double f = ((double)t + 0.5) * 31.25;            // 16000/512 bin width
    double bark = 7.0 * asinh(f / 650.0);
    double maxb = 7.0 * asinh(8000.0 / 650.0);
    double step = maxb / 49.0;
    int bd = (int)(bark / step);
    if (bd > 48) bd = 48;
    if (bd < 0) bd = 0;
    band[t] = bd;
  }
  __syncthreads();
  if (t < NBARK) {
    int cnt = 0;
    for (int k = 0; k < 256; ++k) if (band[k] == t) cnt++;
    double c = (cnt < 1) ? 1.0 : (double)cnt;
    pc[t] = (float)(6.910853e-6 * 100.0 / c);        // SP * 100 / cnt
    double maxb = 7.0 * asinh(8000.0 / 650.0);
    double step = maxb / 49.0;
    double e0 = step * t, e1 = step * (t + 1);
    double fc = 650.0 * sinh(0.5 * (e0 + e1) / 7.0);
    if (fc < 20.0) fc = 20.0;
    fc /= 1000.0;
    double ath = 3.64 * pow(fc, -0.8) - 6.5 * exp(-0.6 * (fc - 3.3) * (fc - 3.3))
               + 1e-3 * pow(fc, 4.0);
    double tv = pow(10.0, ath / 10.0);
    th[t] = (float)tv;
    gS[t] = (float)(1.866055e-1 * pow(tv / 0.5, 0.23));   // SL * (th/0.5)^0.23
  }
}

// DFT matrix pre-swizzled into WMMA B-fragment order:
// entry e = ((ctile*128 + kb)*32 + lane); holds B[k][col], B[k+1][col] with
// k = kb*4 + 2*(lane>>4), col = ctile*16 + (lane&15).
__global__ void k_init_dft(v2f* __restrict__ Bsw) {
  int e = blockIdx.x * 256 + threadIdx.x;   // 0 .. 131071
  int lane = e & 31;
  int kb = (e >> 5) & 127;
  int ctile = e >> 12;
  int k = kb * 4 + 2 * (lane >> 4);
  int col = ctile * 16 + (lane & 15);
  v2f v;
  if (col < 510) {
    int bin = (col >> 1) + 1;                      // bins 1..255
    double w = 2.0 * PI_D * (double)bin / 512.0;
    if ((col & 1) == 0) { v.x = (float)cos(w * (double)k); v.y = (float)cos(w * (double)(k + 1)); }
    else                { v.x = (float)sin(w * (double)k); v.y = (float)sin(w * (double)(k + 1)); }
  } else { v.x = 0.f; v.y = 0.f; }
  Bsw[e] = v;
}

// ----------------------------------------------------- scale + edge taper ---
__global__ void k_prep(const float* __restrict__ in_deg, const float* __restrict__ in_ref,
                       float* __restrict__ X) {      // X = [2][NB][T]: sig0=ref, sig1=deg
  int i = blockIdx.x * 256 + threadIdx.x;
  if (i >= NB * T_SAMP) return;
  int t = i % T_SAMP;
  float tap = 1.f;
  if (t < 15) tap = (float)(t + 1) * (1.f / 16.f);
  else if (t >= T_SAMP - 15) tap = (float)(T_SAMP - t) * (1.f / 16.f);
  X[i]                      = (in_deg[i] * 32320.01953125f) * tap; // "ref" (args swapped)
  X[(size_t)NB * T_SAMP + i] = (in_ref[i] * 23455.2265625f) * tap; // "deg"
}

// ---------------------------------------------------- IIR: chunked scan -----
__global__ void k_iir_chunk(const float* __restrict__ X, float2* __restrict__ V) {
  int id = blockIdx.x * 256 + threadIdx.x;      // 2*NB*NCH
  if (id >= 2 * NB * NCH) return;
  const float* x = X + (size_t)id * LCH;        // [sig][b][chunk] contiguous
  float s1 = 0.f, s2 = 0.f;
  for (int i = 0; i < LCH; ++i) {
    float xv = x[i];
    float y  = C_B0 * xv + s1;
    float n1 = C_B1 * xv - C_A1 * y + s2;
    float n2 = C_B2 * xv - C_A2 * y;
    s1 = n1; s2 = n2;
  }
  V[id] = make_float2(s1, s2);
}

__global__ void k_iir_combine(const float2* __restrict__ V, float2* __restrict__ S) {
  int id = threadIdx.x;                          // 2*NB chains
  if (id >= 2 * NB) return;
  const float a00 = -C_A1, a01 = 1.f, a10 = -C_A2;
  float m00 = 1.f, m01 = 0.f, m10 = 0.f, m11 = 1.f;   // M = A^LCH
  for (int i = 0; i < LCH; ++i) {
    float n00 = a00 * m00 + a01 * m10;
    float n01 = a00 * m01 + a01 * m11;
    float n10 = a10 * m00;
    float n11 = a10 * m01;
    m00 = n00; m01 = n01; m10 = n10; m11 = n11;
  }
  float s1 = 0.f, s2 = 0.f;
  int base = id * NCH;
  for (int k = 0; k < NCH; ++k) {
    S[base + k] = make_float2(s1, s2);
    float2 v = V[base + k];
    float t1 = m00 * s1 + m01 * s2 + v.x;
    float t2 = m10 * s1 + m11 * s2 + v.y;
    s1 = t1; s2 = t2;
  }
}

__global__ void k_iir_apply(float* __restrict__ X, const float2* __restrict__ S) {
  int id = blockIdx.x * 256 + threadIdx.x;
  if (id >= 2 * NB * NCH) return;
  float* x = X + (size_t)id * LCH;
  float2 s = S[id];
  float s1 = s.x, s2 = s.y;
  for (int i = 0; i < LCH; ++i) {
    float xv = x[i];
    float y  = C_B0 * xv + s1;
    float n1 = C_B1 * xv - C_A1 * y + s2;
    float n2 = C_B2 * xv - C_A2 * y;
    s1 = n1; s2 = n2;
    x[i] = y;                                    // in-place within own chunk
  }
}

// -------------------- spectrogram (WMMA fp32 DFT) + power + Bark, fused -----
__global__ __launch_bounds__(256) void k_spect(
    const float* __restrict__ X, const v2f* __restrict__ Bsw,
    const float* __restrict__ win, const int* __restrict__ band,
    const float* __restrict__ pc, float* __restrict__ PB) {
  __shared__ float As[16 * 516];       // 16 frames x 512 (+4 pad)
  __shared__ float Cs[8 * 256];        // per-wave 16x16 C tile
  __shared__ float bk[16 * NBARK];     // Bark accumulators
  int fb = blockIdx.x;                 // 0..2495  (39 blocks per batch row)
  int sig = blockIdx.y;                // 0=ref, 1=deg
  int tid = threadIdx.x;
  int batch = fb / 39;
  int f0 = (fb % 39) * 16;
  const float* Y = X + (size_t)sig * NB * T_SAMP + (size_t)batch * T_SAMP;

  for (int idx = tid; idx < 16 * 512; idx += 256) {
    int m = idx >> 9, j = idx & 511;
    int s = (f0 + m) * 256 + j;
    if (sig) { s = s + 1; if (s > T_SAMP - 1) s = T_SAMP - 1; }  // deg left-shift
    As[m * 516 + j] = Y[s] * win[j];
  }
  for (int idx = tid; idx < 16 * NBARK; idx += 256) bk[idx] = 0.f;
  __syncthreads();

  int wave = tid >> 5, lane = tid & 31;
  int abase = (lane & 15) * 516 + 2 * (lane >> 4);
  for (int cp = 0; cp < 4; ++cp) {
    int ctile = cp * 8 + wave;
    v8f acc = {0.f, 0.f, 0.f, 0.f, 0.f, 0.f, 0.f, 0.f};
    const v2f* bp = Bsw + (size_t)(ctile * 128) * 32 + lane;
    for (int kb = 0; kb < 128; ++kb) {
      v2f a  = *(const v2f*)(&As[abase + kb * 4]);   // A frag: ds_load_b64
      v2f bf = bp[(size_t)kb * 32];                  // B frag: coalesced b64
      acc = __builtin_amdgcn_wmma_f32_16x16x4_f32(false, a, false, bf,
                                                  (short)0, acc, false, false);
    }
    float* cw = &Cs[wave * 256];
    int crow = (lane >> 4) * 8, cn = lane & 15;
#pragma unroll
    for (int r = 0; r < 8; ++r) cw[(crow + r) * 16 + cn] = acc[r];
    __syncthreads();
    int colbase = ctile * 16;
#pragma unroll
    for (int q = 0; q < 4; ++q) {
      int p = q * 32 + lane;                // 128 (frame,bin) pairs per tile
      int m = p >> 3, i = p & 7;
      int col = colbase + 2 * i;
      if (col < 510) {
        float re = cw[m * 16 + 2 * i], im = cw[m * 16 + 2 * i + 1];
        atomicAdd(&bk[m * NBARK + band[(col >> 1) + 1]], re * re + im * im);
      }
    }
    __syncthreads();
  }
  for (int idx = tid; idx < 16 * NBARK; idx += 256) {
    int m = idx / NBARK, n = idx % NBARK;
    PB[(((size_t)sig * NB + batch) * NFRAMES + (f0 + m)) * NBARK + n] = bk[idx] * pc[n];
  }
}

// ------------------------------------------------- perceptual model ---------
__global__ void k_fstats(const float* __restrict__ PB, const float* __restrict__ th,
                         float* __restrict__ silent, float* __restrict__ tadeg) {
  int i = blockIdx.x * 256 + threadIdx.x;
  if (i >= TOTF) return;
  const float* R = PB + (size_t)i * NBARK;
  const float* D = PB + (size_t)(TOTF + i) * NBARK;
  float ta100 = 0.f, ta1 = 0.f;
  for (int n = 1; n < NBARK; ++n) {
    float t = th[n], r = R[n], d = D[n];
    if (r > t * 100.f) ta100 += r;
    if (d > t)         ta1   += d;
  }
  silent[i] = (ta100 < 1e7f) ? 1.f : 0.f;
  tadeg[i] = ta1;
}

__global__ void k_bandmeans(const float* __restrict__ PB, const float* __restrict__ th,
                            const float* __restrict__ silent, float* __restrict__ bpr) {
  int b = blockIdx.x, n = threadIdx.x;
  if (n >= NBARK) return;
  float t100 = th[n] * 100.f;
  float sd = 0.f, sr = 0.f; int cnt = 0;
  for (int f = 0; f < NFRAMES; ++f) {
    int i = b * NFRAMES + f;
    if (silent[i] == 0.f) {
      cnt++;
      float r = PB[(size_t)i * NBARK + n];
      float d = PB[(size_t)(TOTF + i) * NBARK + n];
      if (d > t100) sd += d;
      if (r > t100) sr += r;
    }
  }
  float c = (float)(cnt < 1 ? 1 : cnt);
  float v = (sd / c + 1000.f) / (sr / c + 1000.f);
  bpr[b * NBARK + n] = fminf(fmaxf(v, 0.01f), 100.f);
}

__global__ void k_fpr(const float* __restrict__ PB, const float* __restrict__ th,
                      const float* __restrict__ bpr, const float* __restrict__ tadeg,
                      float* __restrict__ taequ, float* __restrict__ fprraw) {
  int i = blockIdx.x * 256 + threadIdx.x;
  if (i >= TOTF) return;
  int b = i / NFRAMES;
  const float* R = PB + (size_t)i * NBARK;
  float s = 0.f;
  for (int n = 1; n < NBARK; ++n) {
    float er = bpr[b * NBARK + n] * R[n];
    if (er > th[n]) s += er;
  }
  taequ[i] = s;
  fprraw[i] = (s + 5000.f) / (tadeg[i] + 5000.f);
}

__global__ void k_dist(const float* __restrict__ PB, const float* __restrict__ th,
                       const float* __restrict__ gS, const float* __restrict__ bpr,
                       const float* __restrict__ fprraw, const float* __restrict__ taequ,
                       float* __restrict__ symm, float* __restrict__ asym) {
  int i = blockIdx.x * 256 + threadIdx.x;
  if (i >= TOTF) return;
  int b = i / NFRAMES, f = i % NFRAMES;
  float fpr = fprraw[i];
  if (f > 0) fpr = 0.8f * fpr + 0.2f * fprraw[i - 1];    // non-recursive FIR
  fpr = fminf(fmaxf(fpr, 0.0003f), 5.f);
  const float* R = PB + (size_t)i * NBARK;
  const float* D = PB + (size_t)(TOTF + i) * NBARK;
  const float maxbark = 7.f * asinhf(8000.f / 650.f);
  const float w = maxbark / 49.f, wsum = maxbark;
  float acc2 = 0.f, acc1 = 0.f;
  for (int n = 0; n < NBARK; ++n) {
    float t = th[n];
    float er = bpr[b * NBARK + n] * R[n];
    float ed = fpr * D[n];
    float dl = (ed > t) ? gS[n] * (powf(0.5f + 0.5f * ed / t, 0.23f) - 1.f) : 0.f;
    float rl = (er > t) ? gS[n] * (powf(0.5f + 0.5f * er / t, 0.23f) - 1.f) : 0.f;
    float dz = 0.25f * fminf(dl, rl);
    float d = dl - rl;
    float ad = fmaxf(fabsf(d) - dz, 0.f);
    float xw = ad * w;
    acc2 += xw * xw;
    float as = powf((ed + 50.f) / (er + 50.f), 1.2f);
    as = (as < 3.f) ? 0.f : fminf(as, 12.f);
    acc1 += xw * as;
  }
  float sy = fmaxf(sqrtf(acc2 / wsum) * wsum, 1e-20f);
  float ay = fmaxf(acc1, 1e-20f);
  float h = powf((taequ[i] + 1e5f) * 1e-7f, 0.04f);
  symm[i] = fminf(sy / h, 45.f);
  asym[i] = fminf(ay / h, 45.f);
}

__global__ void k_mos(const float* __restrict__ symm, const float* __restrict__ asym,
                      float* __restrict__ out) {
  int b = blockIdx.x * blockDim.x + threadIdx.x;
  if (b >= NB) return;
  float accS = 0.f, accA = 0.f;
  for (int wi = 0; wi < 61; ++wi) {
    int base = b * NFRAMES + wi * 10;
    float s6 = 0.f, a6 = 0.f;
    for (int j = 0; j < 20; ++j) {
      float v = symm[base + j]; float v2 = v * v; s6 += v2 * v2 * v2;
      float u = asym[base + j]; float u2 = u * u; a6 += u2 * u2 * u2;
    }
    float ps = powf(s6 * (1.f / 20.f), 1.f / 6.f);
    float pa = powf(a6 * (1.f / 20.f), 1.f / 6.f);
    accS += ps * ps; accA += pa * pa;
  }
  float ds = sqrtf(accS / 61.f), da = sqrtf(accA / 61.f);
  float mos = 4.5f - 0.1f * ds - 0.0309f * da;
  mos = 0.999f + 4.f / (1.f + expf(-1.3669f * mos + 3.8224f));
  out[b] = 0.5f * (4.5f - mos);
}

// ---------------------------------------------------------------- launch ----
extern "C" void kernel_launch(void* const* d_in, const int* in_sizes, int n_in,
                              void* d_out, int out_size, void* d_ws, size_t ws_size,
                              hipStream_t stream) {
  (void)in_sizes; (void)n_in; (void)out_size; (void)ws_size;
  const float* in_deg = (const float*)d_in[0];
  const float* in_ref = (const float*)d_in[1];
  float* out = (float*)d_out;
  char* ws = (char*)d_ws;

  const size_t OFF_X   = 0;                                  // 2*NB*T f32
  const size_t OFF_V   = OFF_X   + (size_t)2 * NB * T_SAMP * 4;   // 81,920,000
  const size_t OFF_S   = OFF_V   + (size_t)2 * NB * NCH * 8;
  const size_t OFF_BSW = OFF_S   + (size_t)2 * NB * NCH * 8;
  const size_t OFF_WIN = OFF_BSW + (size_t)512 * 512 * 4;
  const size_t OFF_TAB = OFF_WIN + 2048;                     // band/pc/th/gS
  const size_t OFF_PB  = OFF_TAB + 4096;
  const size_t OFF_SIL = OFF_PB  + (size_t)2 * TOTF * NBARK * 4;
  const size_t OFF_TAD = OFF_SIL + (size_t)TOTF * 4;
  const size_t OFF_TAE = OFF_TAD + (size_t)TOTF * 4;
  const size_t OFF_FPR = OFF_TAE + (size_t)TOTF * 4;
  const size_t OFF_SY  = OFF_FPR + (size_t)TOTF * 4;
  const size_t OFF_AS  = OFF_SY  + (size_t)TOTF * 4;
  const size_t OFF_BPR = OFF_AS  + (size_t)TOTF * 4;

  float*  X     = (float*)(ws + OFF_X);
  float2* V     = (float2*)(ws + OFF_V);
  float2* S     = (float2*)(ws + OFF_S);
  v2f*    Bsw   = (v2f*)(ws + OFF_BSW);
  float*  win   = (float*)(ws + OFF_WIN);
  int*    band  = (int*)(ws + OFF_TAB);
  float*  pc    = (float*)(ws + OFF_TAB + 1024);
  float*  th    = (float*)(ws + OFF_TAB + 1280);
  float*  gS    = (float*)(ws + OFF_TAB + 1536);
  float*  PB    = (float*)(ws + OFF_PB);
  float*  sil   = (float*)(ws + OFF_SIL);
  float*  tadeg = (float*)(ws + OFF_TAD);
  float*  taequ = (float*)(ws + OFF_TAE);
  float*  fprr  = (float*)(ws + OFF_FPR);
  float*  symm  = (float*)(ws + OFF_SY);
  float*  asym  = (float*)(ws + OFF_AS);
  float*  bpr   = (float*)(ws + OFF_BPR);

  k_init_tables<<<1, 512, 0, stream>>>(win, band, pc, th, gS);
  k_init_dft<<<512, 256, 0, stream>>>(Bsw);
  k_prep<<<(NB * T_SAMP + 255) / 256, 256, 0, stream>>>(in_deg, in_ref, X);
  k_iir_chunk<<<(2 * NB * NCH + 255) / 256, 256, 0, stream>>>(X, V);
  k_iir_combine<<<1, 128, 0, stream>>>(V, S);
  k_iir_apply<<<(2 * NB * NCH + 255) / 256, 256, 0, stream>>>(X, S);
  k_spect<<<dim3(NB * 39, 2), 256, 0, stream>>>(X, Bsw, win, band, pc, PB);
  k_fstats<<<(TOTF + 255) / 256, 256, 0, stream>>>(PB, th, sil, tadeg);
  k_bandmeans<<<NB, 64, 0, stream>>>(PB, th, sil, bpr);
  k_fpr<<<(TOTF + 255) / 256, 256, 0, stream>>>(PB, th, bpr, tadeg, taequ, fprr);
  k_dist<<<(TOTF + 255) / 256, 256, 0, stream>>>(PB, th, gS, bpr, fprr, taequ, symm, asym);
  k_mos<<<1, 64, 0, stream>>>(symm, asym, out);
}